// Matcher_38001870635223
// MI455X (gfx1250) — compile-verified
//
#include <hip/hip_runtime.h>
#include <hip/hip_bf16.h>
#include <math.h>

typedef __attribute__((ext_vector_type(2))) float v2f;
typedef __attribute__((ext_vector_type(8))) float v8f;

#define DIM     64
#define KSTEPS  16          // 64 / 4 per-WMMA K
#define NWAVES  8
#define TPB     (NWAVES * 32)

#if __has_builtin(__builtin_amdgcn_wmma_f32_16x16x4_f32)
#define HAVE_WMMA_F32X4 1
#endif

__device__ __forceinline__ float xor16(float v) { return __shfl_xor(v, 16, 32); }

// Per-row stats of  Dm[i,j] = 2 - sqrt(max(||P_i||^2 + ||Q_j||^2 - 2 P_i.Q_j, 0))
// over all j in [0, NQ).  Emits argmax (first occurrence) and
// lp = Dm[i,argmax] - logsumexp_j Dm[i,j] = -log(sum_j exp(Dm[i,j] - max)).
__global__ __launch_bounds__(TPB)
void rowstats_kernel(const float* __restrict__ P, const float* __restrict__ Q,
                     int NQ, int* __restrict__ out_arg, float* __restrict__ out_lp)
{
    __shared__ float s_xn[16];
    __shared__ float s_mx[NWAVES][16];
    __shared__ float s_sm[NWAVES][16];
    __shared__ int   s_ix[NWAVES][16];

    const int rowbase = blockIdx.x * 16;
    const int tid  = threadIdx.x;
    const int lane = tid & 31;
    const int wave = tid >> 5;
    const int half = lane >> 4;    // 0: lanes 0-15, 1: lanes 16-31
    const int lm   = lane & 15;

    // Row norms for the 16 P rows this block owns.
    if (tid < 16) {
        const float* pr = P + (size_t)(rowbase + tid) * DIM;
        float s = 0.f;
        #pragma unroll
        for (int k = 0; k < DIM; ++k) s = fmaf(pr[k], pr[k], s);
        s_xn[tid] = s;
    }
    __syncthreads();

    // A fragment (16x4 f32 per step, ISA layout): lane holds row lm,
    // VGPR pair = elements {4k + 2*half, 4k + 2*half + 1}.  Reused for all tiles.
    const float* prow = P + (size_t)(rowbase + lm) * DIM + 2 * half;
    v2f a[KSTEPS];
    #pragma unroll
    for (int kk = 0; kk < KSTEPS; ++kk)
        a[kk] = *(const v2f*)(prow + 4 * kk);

    // C/D slot r -> row (r + 8*half); cache those 8 row norms per lane.
    float xr[8];
    #pragma unroll
    for (int r = 0; r < 8; ++r) xr[r] = s_xn[r + 8 * half];

    // Online-softmax accumulators per (slot row, lane column).
    float mx[8], sm[8];
    int   ix[8];
    #pragma unroll
    for (int r = 0; r < 8; ++r) { mx[r] = -INFINITY; sm[r] = 0.f; ix[r] = 0x7fffffff; }

    const int ntiles = NQ / 16;
    for (int jt = wave; jt < ntiles; jt += NWAVES) {
        const int jbase = jt * 16;

        // B fragment (4x16 f32 per step): identical striping, from Q rows.
        const float* qrow = Q + (size_t)(jbase + lm) * DIM + 2 * half;
        v2f  b[KSTEPS];
        float ynp = 0.f;
        #pragma unroll
        for (int kk = 0; kk < KSTEPS; ++kk) {
            b[kk] = *(const v2f*)(qrow + 4 * kk);
            ynp = fmaf(b[kk].x, b[kk].x, ynp);
            ynp = fmaf(b[kk].y, b[kk].y, ynp);
        }
        const float ynorm = ynp + xor16(ynp);   // ||Q_{jbase+lm}||^2 on both halves

#ifdef HAVE_WMMA_F32X4
        v8f c = {};
        #pragma unroll
        for (int kk = 0; kk < KSTEPS; ++kk)
            c = __builtin_amdgcn_wmma_f32_16x16x4_f32(false, a[kk], false, b[kk],
                                                      (short)0, c, false, false);
#else
        // Fallback (compile-feedback only): scalar dot products.
        v8f c = {};
        {
            const float* qc = Q + (size_t)(jbase + lm) * DIM;
            #pragma unroll
            for (int r = 0; r < 8; ++r) {
                const float* pr = P + (size_t)(rowbase + r + 8 * half) * DIM;
                float acc = 0.f;
                for (int k = 0; k < DIM; ++k) acc = fmaf(pr[k], qc[k], acc);
                c[r] = acc;
            }
        }
#endif
        const int col = jbase + lm;
        #pragma unroll
        for (int r = 0; r < 8; ++r) {
            float d2 = c[r];
            float sq = fmaf(-2.f, d2, xr[r] + ynorm);
            sq = fmaxf(sq, 0.f);
            float dm = 2.f - sqrtf(sq);
            float nm = fmaxf(mx[r], dm);
            sm[r] = fmaf(sm[r], __expf(mx[r] - nm), __expf(dm - nm));
            ix[r] = (dm > mx[r]) ? col : ix[r];   // cols strictly increase per lane
            mx[r] = nm;
        }
    }

    // Reduce the 16 columns (within each half-wave; halves are different rows).
    #pragma unroll
    for (int r = 0; r < 8; ++r) {
        #pragma unroll
        for (int dlt = 1; dlt < 16; dlt <<= 1) {
            float omx = __shfl_xor(mx[r], dlt, 32);
            float osm = __shfl_xor(sm[r], dlt, 32);
            int   oix = __shfl_xor(ix[r], dlt, 32);
            float nm  = fmaxf(mx[r], omx);
            sm[r] = fmaf(sm[r], __expf(mx[r] - nm), osm * __expf(omx - nm));
            bool take = (omx > mx[r]) || ((omx == mx[r]) && (oix < ix[r]));
            ix[r] = take ? oix : ix[r];
            mx[r] = nm;
        }
    }
    if (lm == 0) {
        #pragma unroll
        for (int r = 0; r < 8; ++r) {
            const int row = r + 8 * half;
            s_mx[wave][row] = mx[r];
            s_sm[wave][row] = sm[r];
            s_ix[wave][row] = ix[r];
        }
    }
    __syncthreads();

    // Merge the 8 wave partials per row; emit argmax and -log(sum).
    if (tid < 16) {
        float M = -INFINITY, S = 0.f;
        int   I = 0x7fffffff;
        #pragma unroll
        for (int w = 0; w < NWAVES; ++w) {
            float m = s_mx[w][tid], s = s_sm[w][tid];
            int   i = s_ix[w][tid];
            float nm = fmaxf(M, m);
            S = fmaf(S, __expf(M - nm), s * __expf(m - nm));
            bool take = (m > M) || ((m == M) && (i < I));
            I = take ? i : I;
            M = nm;
        }
        out_arg[rowbase + tid] = I;
        out_lp[rowbase + tid]  = -__logf(S);
    }
}

// Mutual-match finalize: out = [log_probs(M) | dmatches(M,2) | mutual(M)], all f32.
__global__ void finalize_kernel(const int* __restrict__ row_arg,
                                const float* __restrict__ row_lp,
                                const int* __restrict__ col_arg,
                                const float* __restrict__ col_lp,
                                float* __restrict__ out, int M)
{
    int j = blockIdx.x * blockDim.x + threadIdx.x;
    if (j >= M) return;
    int  a   = col_arg[j];
    bool mut = (row_arg[a] == j);
    out[j]             = mut ? (row_lp[a] + col_lp[j]) : 0.f;
    out[M + 2 * j]     = (float)a;
    out[M + 2 * j + 1] = (float)j;
    out[3 * M + j]     = mut ? 1.f : 0.f;
}

extern "C" void kernel_launch(void* const* d_in, const int* in_sizes, int n_in,
                              void* d_out, int out_size, void* d_ws, size_t ws_size,
                              hipStream_t stream) {
    const float* x = (const float*)d_in[0];   // [N, 64] f32
    const float* y = (const float*)d_in[1];   // [M, 64] f32
    const int N = in_sizes[0] / DIM;          // 8192
    const int M = in_sizes[1] / DIM;          // 8192

    char* ws = (char*)d_ws;
    int*   row_arg = (int*)  (ws);
    float* row_lp  = (float*)(ws + (size_t)N * 4);
    int*   col_arg = (int*)  (ws + (size_t)N * 8);
    float* col_lp  = (float*)(ws + (size_t)N * 8 + (size_t)M * 4);

    // Row stats of Dmat (P = x, columns over y), then of Dmat^T (P = y, columns over x).
    rowstats_kernel<<<dim3(N / 16), dim3(TPB), 0, stream>>>(x, y, M, row_arg, row_lp);
    rowstats_kernel<<<dim3(M / 16), dim3(TPB), 0, stream>>>(y, x, N, col_arg, col_lp);

    finalize_kernel<<<dim3((M + 255) / 256), dim3(256), 0, stream>>>(
        row_arg, row_lp, col_arg, col_lp, (float*)d_out, M);
}